// NeuralDictionaryV10_38594576121965
// MI455X (gfx1250) — compile-verified
//
#include <hip/hip_runtime.h>
#include <hip/hip_bf16.h>

// N = 65536 keys, D = 256, DV = 256, all fp32.
// d_in[0] = query (256 f32), d_in[1] = keys (65536x256 f32), d_in[2] = values (65536x256 f32)
// d_out = 256 f32.
//
// ws layout (floats):
//   [0      .. 65536)  logits
//   [65536  .. 65792)  per-block maxes (256)
//   [65792]            global max
//   [65824  .. 66080)  per-block partial sum-of-weights (256)
//   [66080  .. 131616) per-block partial outputs (256 blocks x 256 cols)

typedef __attribute__((ext_vector_type(2))) float v2f;
typedef __attribute__((ext_vector_type(8))) float v8f;

#define NKEYS   65536
#define DIM     256
#define ATT_BLOCKS 256           // blocks in attend kernel
#define KEYS_PER_ATT_BLOCK (NKEYS / ATT_BLOCKS)  // 256

#define WS_LOGITS 0
#define WS_BMAX   65536
#define WS_GMAX   65792
#define WS_PSUM   65824
#define WS_POUT   66080

// ---------------------------------------------------------------------------
// Kernel 1: per-key logit = -sum_d |K[n,d] - q[d]|. One wave (32 lanes) per key,
// 8 floats (2 x float4) per lane, then a 5-step shfl_xor wave reduction.
// ---------------------------------------------------------------------------
__global__ void nd_logits_kernel(const float* __restrict__ query,
                                 const float* __restrict__ keys,
                                 float* __restrict__ logits) {
    const int lane = threadIdx.x & 31;
    const int wave = threadIdx.x >> 5;
    const int key  = blockIdx.x * 8 + wave;   // 8 waves per 256-thread block

    const float4* __restrict__ K4 = (const float4*)(keys + (size_t)key * DIM);
    const float4* __restrict__ Q4 = (const float4*)query;

    float s = 0.0f;
#pragma unroll
    for (int i = 0; i < 2; ++i) {
        const int idx = i * 32 + lane;        // contiguous across lanes
        float4 kv = K4[idx];
        float4 qv = Q4[idx];
        s += fabsf(kv.x - qv.x) + fabsf(kv.y - qv.y) +
             fabsf(kv.z - qv.z) + fabsf(kv.w - qv.w);
    }
#pragma unroll
    for (int off = 16; off > 0; off >>= 1)
        s += __shfl_xor(s, off, 32);

    if (lane == 0) logits[key] = -s;
}

// ---------------------------------------------------------------------------
// Kernel 2a/2b: two-level deterministic max reduction over the logits.
// ---------------------------------------------------------------------------
__global__ void nd_bmax_kernel(const float* __restrict__ logits,
                               float* __restrict__ bmax) {
    __shared__ float red[256];
    const int t = threadIdx.x;
    red[t] = logits[blockIdx.x * 256 + t];
    __syncthreads();
#pragma unroll
    for (int s = 128; s > 0; s >>= 1) {
        if (t < s) red[t] = fmaxf(red[t], red[t + s]);
        __syncthreads();
    }
    if (t == 0) bmax[blockIdx.x] = red[0];
}

__global__ void nd_gmax_kernel(const float* __restrict__ bmax,
                               float* __restrict__ gmax) {
    __shared__ float red[256];
    const int t = threadIdx.x;
    red[t] = bmax[t];
    __syncthreads();
#pragma unroll
    for (int s = 128; s > 0; s >>= 1) {
        if (t < s) red[t] = fmaxf(red[t], red[t + s]);
        __syncthreads();
    }
    if (t == 0) gmax[0] = red[0];
}

// ---------------------------------------------------------------------------
// Kernel 3: per 256-key block: weights into LDS (+ deterministic tree sum),
// then stream V through V_WMMA_F32_16X16X4_F32.
//   A (16x4 f32, 2 VGPRs): lanes 0-15 hold {K0,K1}, lanes 16-31 hold {K2,K3};
//     we broadcast the 4 chunk weights down the M dimension.
//   B (4x16 f32, 2 VGPRs): VGPR0 = rows K0(lanes 0-15)/K1(lanes 16-31),
//     VGPR1 = rows K2/K3 -> natural row-major V loads, 64B per half-wave.
//   D rows are all identical = partial out for that 16-column tile.
// Each of the 8 waves owns two 16-column tiles (8*2*16 = 256 = DV).
// All V addressing is 32-bit element indices (64 MB buffer => byte offsets
// fit in i32) so loads lower to SADDR-base + VGPR-offset `scale_offset` form
// with one 32-bit add per chunk instead of 64-bit pointer arithmetic.
// ---------------------------------------------------------------------------
__global__ void nd_attend_kernel(const float* __restrict__ logits,
                                 const float* __restrict__ gmax,
                                 const float* __restrict__ values,
                                 float* __restrict__ pout,
                                 float* __restrict__ psum) {
    __shared__ float wsm[256];
    __shared__ float red[256];

    const int t    = threadIdx.x;
    const int base = blockIdx.x * KEYS_PER_ATT_BLOCK;

    // Phase A: softmax weights for this block's keys + partial sum.
    const float m = gmax[0];
    const float w = __expf(logits[base + t] - m);
    wsm[t] = w;
    red[t] = w;
    __syncthreads();
#pragma unroll
    for (int s = 128; s > 0; s >>= 1) {
        if (t < s) red[t] += red[t + s];
        __syncthreads();
    }
    if (t == 0) psum[blockIdx.x] = red[0];

    // Phase B: WMMA-accumulated weighted sum of values.
    const int wave = t >> 5;
    const int lane = t & 31;
    const int half = lane >> 4;     // 0: lanes 0-15, 1: lanes 16-31
    const int nl   = lane & 15;
    const int cb0  = wave * 32;     // first 16-col tile for this wave
    const int cb1  = wave * 32 + 16;

    v8f acc0 = {};
    v8f acc1 = {};

    // 32-bit element index of this lane's slot in rows {k0/k1} (by half).
    unsigned int idx01 = (unsigned int)(base + half) * DIM + (unsigned int)nl;
    const unsigned int STEP = 4u * DIM;          // 4 key rows per chunk
    const unsigned int PF   = 8u * STEP;         // prefetch 8 chunks (~32KB) ahead

    for (int c = 0; c < KEYS_PER_ATT_BLOCK / 4; ++c) {
        const float w0 = wsm[c * 4 + 0];
        const float w1 = wsm[c * 4 + 1];
        const float w2 = wsm[c * 4 + 2];
        const float w3 = wsm[c * 4 + 3];

        v2f a;
        a[0] = half ? w2 : w0;      // A VGPR0: K0 (lo half) / K2 (hi half)
        a[1] = half ? w3 : w1;      // A VGPR1: K1 (lo half) / K3 (hi half)

        const unsigned int idx23 = idx01 + 2u * DIM;   // rows k2/k3 (by half)

        // Speculative prefetch of the stream ahead (global_prefetch_b8);
        // dropped silently if past the buffer.
        __builtin_prefetch(values + (idx01 + PF), 0, 1);

        v2f b0, b1;
        b0[0] = values[idx01 + (unsigned int)cb0];  // B VGPR0: row K0 (lo) / K1 (hi)
        b0[1] = values[idx23 + (unsigned int)cb0];  // B VGPR1: row K2 (lo) / K3 (hi)
        b1[0] = values[idx01 + (unsigned int)cb1];
        b1[1] = values[idx23 + (unsigned int)cb1];

        acc0 = __builtin_amdgcn_wmma_f32_16x16x4_f32(
            false, a, false, b0, (short)0, acc0, false, false);
        acc1 = __builtin_amdgcn_wmma_f32_16x16x4_f32(
            false, a, false, b1, (short)0, acc1, false, false);

        idx01 += STEP;
    }

    // D layout: VGPR0 lanes 0-15 = (M=0, N=lane); rows are replicated so this
    // is the partial output for the tile's 16 columns.
    if (lane < 16) {
        pout[blockIdx.x * DIM + cb0 + nl] = acc0[0];
        pout[blockIdx.x * DIM + cb1 + nl] = acc1[0];
    }
}

// ---------------------------------------------------------------------------
// Kernel 4: fixed-order reduction of the 256 block partials, divide by sum(w).
// ---------------------------------------------------------------------------
__global__ void nd_finalize_kernel(const float* __restrict__ pout,
                                   const float* __restrict__ psum,
                                   float* __restrict__ out) {
    const int j = threadIdx.x;
    float s = 0.0f;
    for (int b = 0; b < ATT_BLOCKS; ++b) s += psum[b];
    float acc = 0.0f;
    for (int b = 0; b < ATT_BLOCKS; ++b) acc += pout[b * DIM + j];
    out[j] = acc / s;
}

extern "C" void kernel_launch(void* const* d_in, const int* in_sizes, int n_in,
                              void* d_out, int out_size, void* d_ws, size_t ws_size,
                              hipStream_t stream) {
    const float* query  = (const float*)d_in[0];
    const float* keys   = (const float*)d_in[1];
    const float* values = (const float*)d_in[2];
    float*       out    = (float*)d_out;
    float*       ws     = (float*)d_ws;

    float* logits = ws + WS_LOGITS;
    float* bmax   = ws + WS_BMAX;
    float* gmax   = ws + WS_GMAX;
    float* psum   = ws + WS_PSUM;
    float* pout   = ws + WS_POUT;

    nd_logits_kernel<<<NKEYS / 8, 256, 0, stream>>>(query, keys, logits);
    nd_bmax_kernel<<<NKEYS / 256, 256, 0, stream>>>(logits, bmax);
    nd_gmax_kernel<<<1, 256, 0, stream>>>(bmax, gmax);
    nd_attend_kernel<<<ATT_BLOCKS, 256, 0, stream>>>(logits, gmax, values, pout, psum);
    nd_finalize_kernel<<<1, 256, 0, stream>>>(pout, psum, out);
}